// CNF_7095285973213
// MI455X (gfx1250) — compile-verified
//
#include <hip/hip_runtime.h>
#include <math.h>

typedef __attribute__((ext_vector_type(2))) float v2f;
typedef __attribute__((ext_vector_type(8))) float v8f;

#define D_IN   64
#define WID    256
#define PITCH  260          // 256 + 4 pad to break LDS bank alignment
#define NSTAGE 6
#define NSTEP  10
#define SMP    2            // samples per workgroup
#define NCOL   130          // 128 Jacobian columns + 2 activation columns
#define NTHR   512          // 16 wave32

// Tsit5 tableau
__constant__ float c_A[6][5] = {
  {0.f,0.f,0.f,0.f,0.f},
  {0.161f,0.f,0.f,0.f,0.f},
  {-0.008480655492356989f, 0.335480655492357f, 0.f,0.f,0.f},
  {2.8971530571054935f, -6.359448489975075f, 4.3622954328695815f, 0.f,0.f},
  {5.325864828439257f, -11.748883564062828f, 7.4955393428898365f, -0.09249506636175525f, 0.f},
  {5.86145544294642f, -12.92096931784711f, 8.159367898576159f, -0.071584973281401f, -0.028269050394068383f},
};
__constant__ float c_C[6] = {0.f, 0.161f, 0.327f, 0.9f, 0.9800255409045097f, 1.f};
__constant__ float c_B[6] = {0.09646076681806523f, 0.01f, 0.4798896504144996f,
                             1.379008574103742f, -3.290069515436081f, 2.324710524099774f};

struct CnfParams {
  const float* W1[2][4];
  const float* W2[2][4];
  const float* W3[2][4];
  const float* b1[2][4];
  const float* b2[2][4];
};

static __device__ __forceinline__ v8f wmma_f32_4(v2f a, v2f b, v8f c) {
  // V_WMMA_F32_16X16X4_F32 : D(16x16,f32) = A(16x4,f32) * B(4x16,f32) + C
  return __builtin_amdgcn_wmma_f32_16x16x4_f32(false, a, false, b, (short)0, c, false, false);
}

__global__ __launch_bounds__(NTHR)
void cnf_tsit5_kernel(const float* __restrict__ yin, float* __restrict__ out, CnfParams P)
{
  // Nt[col][k]: cols 0..63 = sample0 Jacobian, 64..127 = sample1 Jacobian,
  //             col 128/129 = forward activation vector of sample 0/1
  __shared__ float Nt[NCOL * PITCH];
  __shared__ float glds[4][WID];            // per-stage gates  (t-only, sample-independent)
  __shared__ float slds[4][WID];            // per-stage shifts (t-only)
  __shared__ float rl[SMP][WID];            // current-layer r = (1-h^2)*gate
  __shared__ float ybuf[SMP][D_IN];         // stage input yi
  __shared__ float ksy[SMP][NSTAGE][D_IN];  // RK stage derivatives (y)
  __shared__ float ksl[SMP][NSTAGE];        // RK stage derivatives (logdet)
  __shared__ float ys[SMP][D_IN];           // current y
  __shared__ float redt[SMP][256];          // trace partials
  __shared__ float redf[SMP][256];          // layer-3 forward partials
  __shared__ float lv[SMP];                 // logdet accumulators

  const int tid  = threadIdx.x;
  const int lane = tid & 31;
  const int wv   = tid >> 5;                // 16 wave32
  const int m16  = lane & 15;
  const int khalf = (lane >> 4) << 1;       // K sub-pair per ISA 16x4 fp32 layout
  const int mofs  = (lane >> 4) << 3;       // D rows +8 for upper half-wave
  const int fs = tid & 1;                   // (row, sample) mapping: adjacent lanes same row
  const int fr = tid >> 1;
  const int rsmp = tid >> 8;                // reduction mapping: block halves = samples
  const int r8   = tid & 255;
  const float dt = -0.05f;                  // integrate t1 -> t0

  if (tid < SMP * D_IN) ys[fs][fr] = yin[(blockIdx.x * SMP + fs) * D_IN + fr];
  if (tid < SMP) lv[tid] = 0.f;
  __syncthreads();

  for (int bb = 0; bb < 2; ++bb) {
    const int blk = 1 - bb;                 // reversed(params)
    for (int step = 0; step < NSTEP; ++step) {
      const float tbase = 0.5f + dt * (float)step;
      for (int st = 0; st < NSTAGE; ++st) {
        const float ti = tbase + c_C[st] * dt;

        // ------------- per-stage gates/shifts (t-only) + stage input -------------
        if (tid < WID) {
          #pragma unroll
          for (int L = 0; L < 4; ++L) {
            if (tid < ((L == 3) ? D_IN : WID)) {
              glds[L][tid] = 1.f / (1.f + __expf(-(P.W2[blk][L][tid] * ti + P.b2[blk][L][tid])));
              slds[L][tid] = P.W3[blk][L][tid] * ti;
            }
          }
        }
        if (tid < SMP * D_IN) {
          float v = ys[fs][fr];
          for (int j = 0; j < st; ++j) v += (dt * c_A[st][j]) * ksy[fs][j][fr];
          ybuf[fs][fr] = v;
        }
        __syncthreads();

        // ------------- layer 0 fused: h0 + N0 = diag(r0)*W0 (one W0 read) -------------
        {
          const int k = fr;                                   // 0..255
          const float* w0row = P.W1[blk][0] + k * D_IN;       // 256x64 row-major
          const float4* w4 = (const float4*)w0row;
          const float4* y4 = (const float4*)(&ybuf[fs][0]);
          float u = P.b1[blk][0][k];
          #pragma unroll
          for (int i = 0; i < 16; ++i) {
            const float4 w = w4[i];
            const float4 y = y4[i];
            u += w.x * y.x + w.y * y.y + w.z * y.z + w.w * y.w;
          }
          const float h0 = tanhf(u * glds[0][k] + slds[0][k]);
          const float r0 = (1.f - h0 * h0) * glds[0][k];
          for (int c = 0; c < D_IN; ++c)                       // 2nd read hits WGP$
            Nt[(fs * D_IN + c) * PITCH + k] = r0 * w0row[c];
          Nt[(128 + fs) * PITCH + k] = h0;
        }
        __syncthreads();

        // ------------- layers 1,2: fused Jacobian GEMM + forward column -------------
        for (int l = 1; l <= 2; ++l) {
          const float* W = P.W1[blk][l];      // 256x256; each row read by ONE wave
          const int rt = wv;

          v8f acc[9];
          #pragma unroll
          for (int c = 0; c < 9; ++c)
            #pragma unroll
            for (int v = 0; v < 8; ++v) acc[c][v] = 0.f;

          const float* arow = W + (rt * 16 + m16) * WID;
          const float* bp[9];
          #pragma unroll
          for (int ct = 0; ct < 8; ++ct) bp[ct] = &Nt[(ct * 16 + m16) * PITCH];
          // tile 8: 2 real activation columns; other 14 lanes alias valid memory (outputs unused)
          bp[8] = &Nt[((m16 < 2) ? (128 + m16) : (m16 - 2)) * PITCH];

          for (int kb = 0; kb < 64; ++kb) {
            const int k0 = (kb << 2) + khalf;
            const v2f a = *(const v2f*)(arow + k0);
            #pragma unroll
            for (int ct = 0; ct < 9; ++ct)
              acc[ct] = wmma_f32_4(a, *(const v2f*)(bp[ct] + k0), acc[ct]);
          }
          __syncthreads();   // all waves finished reading Nt

          // forward: lanes holding activation column compute h_l, r_l
          if (m16 < 2) {
            const int smp = m16;
            #pragma unroll
            for (int v = 0; v < 8; ++v) {
              const int row = rt * 16 + mofs + v;
              const float u = acc[8][v] + P.b1[blk][l][row];
              const float h = tanhf(u * glds[l][row] + slds[l][row]);
              Nt[(128 + smp) * PITCH + row] = h;
              rl[smp][row] = (1.f - h * h) * glds[l][row];
            }
          }
          __syncthreads();   // r_l and h_l visible

          // scale Jacobian tiles by r_l and write back transposed
          #pragma unroll
          for (int ct = 0; ct < 8; ++ct) {
            const int col = ct * 16 + m16;
            const float* rs = rl[ct >> 2];    // cols 0..63 -> sample0, 64..127 -> sample1
            #pragma unroll
            for (int v = 0; v < 8; ++v) {
              const int row = rt * 16 + mofs + v;
              Nt[col * PITCH + row] = acc[ct][v] * rs[row];
            }
          }
          __syncthreads();
        }

        // ------------- layer 3 fused: trace + forward k_y (one W3 read) -------------
        {
          const int i  = r8 >> 2;             // output row 0..63
          const int k0 = (r8 & 3) << 6;       // 64-wide partial
          const float4* w3r = (const float4*)(P.W1[blk][3] + i * WID + k0);
          const float4* nr  = (const float4*)(&Nt[(rsmp * D_IN + i) * PITCH + k0]);
          const float4* hr  = (const float4*)(&Nt[(128 + rsmp) * PITCH + k0]);
          float tp = 0.f, fp = 0.f;
          for (int k = 0; k < 16; ++k) {
            const float4 w = w3r[k];
            const float4 n = nr[k];
            const float4 h = hr[k];
            tp += w.x * n.x + w.y * n.y + w.z * n.z + w.w * n.w;
            fp += w.x * h.x + w.y * h.y + w.z * h.z + w.w * h.w;
          }
          redt[rsmp][r8] = tp * glds[3][i];
          redf[rsmp][r8] = fp;
        }
        __syncthreads();
        if ((r8 & 3) == 0) {
          const int i = r8 >> 2;
          const float u = redf[rsmp][r8] + redf[rsmp][r8 + 1] + redf[rsmp][r8 + 2]
                        + redf[rsmp][r8 + 3] + P.b1[blk][3][i];
          ksy[rsmp][st][i] = u * glds[3][i] + slds[3][i];
        }
        for (int off = 128; off > 0; off >>= 1) {
          if (r8 < off) redt[rsmp][r8] += redt[rsmp][r8 + off];
          __syncthreads();
        }
        if (r8 == 0) ksl[rsmp][st] = redt[rsmp][0];
        __syncthreads();
      } // stages

      // ------------- RK combine -------------
      if (tid < SMP * D_IN) {
        float v = ys[fs][fr];
        for (int i = 0; i < NSTAGE; ++i) v += (dt * c_B[i]) * ksy[fs][i][fr];
        ys[fs][fr] = v;
      }
      if (tid < SMP) {
        float v = lv[tid];
        for (int i = 0; i < NSTAGE; ++i) v += (dt * c_B[i]) * ksl[tid][i];
        lv[tid] = v;
      }
      __syncthreads();
    } // steps
  } // blocks

  // ------------- log-likelihood -------------
  redt[rsmp][r8] = (r8 < D_IN) ? ys[rsmp][r8] * ys[rsmp][r8] : 0.f;
  __syncthreads();
  for (int off = 128; off > 0; off >>= 1) {
    if (r8 < off) redt[rsmp][r8] += redt[rsmp][r8 + off];
    __syncthreads();
  }
  if (r8 == 0)
    out[blockIdx.x * SMP + rsmp] = lv[rsmp] - 0.5f * (64.f * 1.8378770664093453f + redt[rsmp][0]);
}

extern "C" void kernel_launch(void* const* d_in, const int* in_sizes, int n_in,
                              void* d_out, int out_size, void* d_ws, size_t ws_size,
                              hipStream_t stream) {
  (void)n_in; (void)out_size; (void)d_ws; (void)ws_size;
  const float* y = (const float*)d_in[0];   // d_in[1] = eps (unused in exact mode)
  CnfParams P;
  int idx = 2;  // pytree order: blocks 0..1, layers 0..3, keys sorted W1,W2,W3,b1,b2
  for (int b = 0; b < 2; ++b) {
    for (int l = 0; l < 4; ++l) {
      P.W1[b][l] = (const float*)d_in[idx + 0];
      P.W2[b][l] = (const float*)d_in[idx + 1];
      P.W3[b][l] = (const float*)d_in[idx + 2];
      P.b1[b][l] = (const float*)d_in[idx + 3];
      P.b2[b][l] = (const float*)d_in[idx + 4];
      idx += 5;
    }
  }
  const int batch = in_sizes[0] / D_IN;     // 2048
  cnf_tsit5_kernel<<<batch / SMP, NTHR, 0, stream>>>(y, (float*)d_out, P);
}